// TPSModule_25744033973207
// MI455X (gfx1250) — compile-verified
//
#include <hip/hip_runtime.h>
#include <math.h>

// TPS token-pruning similarity module, fused for MI455X (gfx1250, wave32).
// sim = xn @ xn^T done with V_WMMA_F32_16X16X4_F32 tiles; the sparse
// "top-1 scatter" phase uses float global atomics; sim is never materialized.

typedef __attribute__((ext_vector_type(2))) float v2f;
typedef __attribute__((ext_vector_type(8))) float v8f;

#define BB    16
#define NN    2048
#define CCH   64
#define TILE  16
#define KCH   16          // 64 / 4 K-chunks per 16x16 tile
#define NCT   (NN / TILE) // 128 column tiles
#define STR   68          // LDS row stride (floats): bank = (4*row + k) % 64, conflict-free
#define NEGV  (-100.0f)
#define EPSV  (1e-6f)
#define EULER 2.71828182845904523536f
#define WAVES 4

__global__ void tps_zero(float* __restrict__ acc, float* __restrict__ w) {
    size_t total = (size_t)BB * NN * CCH;
    size_t stride = (size_t)gridDim.x * blockDim.x;
    for (size_t i = (size_t)blockIdx.x * blockDim.x + threadIdx.x; i < total; i += stride) {
        acc[i] = 0.0f;
        if (i < (size_t)BB * NN) w[i] = 0.0f;
    }
}

__global__ void tps_rnorm(const float* __restrict__ x, float* __restrict__ rn) {
    int t = blockIdx.x * blockDim.x + threadIdx.x;
    if (t >= BB * NN) return;
    const float* p = x + (size_t)t * CCH;
    float s = 0.0f;
#pragma unroll 8
    for (int c = 0; c < CCH; ++c) { float v = p[c]; s += v * v; }
    rn[t] = 1.0f / (sqrtf(s) + EPSV);
}

__launch_bounds__(WAVES * 32)
__global__ void tps_sim(const float* __restrict__ x,
                        const float* __restrict__ keep,
                        const float* __restrict__ pruned,
                        const float* __restrict__ rn,
                        float* __restrict__ w,
                        float* __restrict__ acc) {
    __shared__ __align__(16) float lds[WAVES][2][TILE * STR];

    const int lane = threadIdx.x & 31;
    const int wv   = threadIdx.x >> 5;
    const int tile = blockIdx.x * WAVES + wv;
    const int bidx = tile / (NN / TILE);
    const int r0   = (tile % (NN / TILE)) * TILE;
    const int half = lane >> 4;   // 0: rows r0..r0+7, 1: rows r0+8..r0+15 (C/D layout)
    const int l16  = lane & 15;   // this lane's column within the 16-col tile

    float* As = lds[wv][0];
    float* Bs = lds[wv][1];
    const size_t bbase = (size_t)bidx * NN;

    // ---- stage normalized A tile (rows r0..r0+15) into LDS ----
#pragma unroll
    for (int it = 0; it < 8; ++it) {
        int t   = lane + 32 * it;     // 0..255 float4 slots
        int row = t >> 4;
        int c4  = t & 15;
        int tok = r0 + row;
        float4 v = reinterpret_cast<const float4*>(x + (bbase + tok) * CCH)[c4];
        float  s = rn[bbase + tok];
        v.x *= s; v.y *= s; v.z *= s; v.w *= s;
        *reinterpret_cast<float4*>(As + row * STR + c4 * 4) = v;
    }
    __syncthreads();

    // A fragments in registers: 16x4 f32 layout — lanes 0-15 hold M=lane,
    // VGPR0=K0 / VGPR1=K1; lanes 16-31 hold K2/K3.
    v2f afrag[KCH];
#pragma unroll
    for (int kk = 0; kk < KCH; ++kk) {
        afrag[kk].x = As[l16 * STR + 4 * kk + 2 * half];
        afrag[kk].y = As[l16 * STR + 4 * kk + 2 * half + 1];
    }

    float rowmax[8];
    float prow[8];
#pragma unroll
    for (int v = 0; v < 8; ++v) rowmax[v] = -INFINITY;

    for (int pass = 0; pass < 2; ++pass) {
        for (int ct = 0; ct < NCT; ++ct) {
            const int c0 = ct * TILE;
            __syncthreads();
            // stage normalized B tile (column tokens c0..c0+15)
#pragma unroll
            for (int it = 0; it < 8; ++it) {
                int t   = lane + 32 * it;
                int row = t >> 4;
                int c4  = t & 15;
                int tok = c0 + row;
                float4 v = reinterpret_cast<const float4*>(x + (bbase + tok) * CCH)[c4];
                float  s = rn[bbase + tok];
                v.x *= s; v.y *= s; v.z *= s; v.w *= s;
                *reinterpret_cast<float4*>(Bs + row * STR + c4 * 4) = v;
            }
            __syncthreads();

            const int   col = c0 + l16;
            const float kc  = keep[bbase + col];

            v8f cacc = {};
#pragma unroll
            for (int kk = 0; kk < KCH; ++kk) {
                v2f bf;
                bf.x = Bs[l16 * STR + 4 * kk + 2 * half];
                bf.y = Bs[l16 * STR + 4 * kk + 2 * half + 1];
                // D = A x B + C ; (neg_a, A, neg_b, B, c_mod, C, reuse_a, reuse_b)
                cacc = __builtin_amdgcn_wmma_f32_16x16x4_f32(
                    false, afrag[kk], false, bf, (short)0, cacc, false, false);
            }

#pragma unroll
            for (int v = 0; v < 8; ++v) {
                const int grow = r0 + v + 8 * half;   // source row this element belongs to
                float s = cacc[v];
                if (grow == col)  s = NEGV;           // eye mask
                if (kc == 0.0f)   s = NEGV;           // non-kept column mask
                if (pass == 0) {
                    rowmax[v] = fmaxf(rowmax[v], s);
                } else {
                    // bitwise-identical recompute -> exact tie detection
                    if (prow[v] != 0.0f && s == rowmax[v]) {
                        float e = expf(s);
                        atomicAdd(&w[bbase + col], e);
                        const float* xr = x   + (bbase + grow) * CCH;
                        float*       ar = acc + (bbase + col) * CCH;
#pragma unroll 8
                        for (int c = 0; c < CCH; ++c)
                            atomicAdd(&ar[c], e * xr[c]);
                    }
                }
            }
        }
        if (pass == 0) {
            // reduce row max across the 16 lanes of each half (each lane = 1 column)
#pragma unroll
            for (int v = 0; v < 8; ++v) {
                float m = rowmax[v];
                for (int off = 1; off < 16; off <<= 1)
                    m = fmaxf(m, __shfl_xor(m, off, 32));
                rowmax[v] = m;
                prow[v]   = pruned[bbase + r0 + v + 8 * half];
            }
        }
    }
}

__global__ void tps_final(const float* __restrict__ x,
                          const float* __restrict__ w,
                          float* __restrict__ out) {
    size_t total  = (size_t)BB * NN * CCH;
    size_t stride = (size_t)gridDim.x * blockDim.x;
    for (size_t i = (size_t)blockIdx.x * blockDim.x + threadIdx.x; i < total; i += stride) {
        size_t tok = i / CCH;
        float den = EULER + w[tok];
        out[i] = (x[i] * EULER + out[i]) / den;
    }
}

extern "C" void kernel_launch(void* const* d_in, const int* in_sizes, int n_in,
                              void* d_out, int out_size, void* d_ws, size_t ws_size,
                              hipStream_t stream) {
    const float* x      = (const float*)d_in[0];
    // d_in[1] = y (unused by the reference's training branch)
    const float* keep   = (const float*)d_in[2];
    const float* pruned = (const float*)d_in[3];
    float* out = (float*)d_out;

    float* rn = (float*)d_ws;          // B*N floats  (128 KB)
    float* w  = rn + (size_t)BB * NN;  // B*N floats  (128 KB)

    tps_zero<<<512, 256, 0, stream>>>(out, w);
    tps_rnorm<<<(BB * NN + 255) / 256, 256, 0, stream>>>(x, rn);

    dim3 grid(BB * (NN / TILE) / WAVES);   // 512 blocks x 4 waves = 2048 row tiles
    tps_sim<<<grid, WAVES * 32, 0, stream>>>(x, keep, pruned, rn, w, out);

    tps_final<<<512, 256, 0, stream>>>(x, w, out);
}